// CausalSelfAttention_63737314673194
// MI455X (gfx1250) — compile-verified
//
#include <hip/hip_runtime.h>
#include <hip/hip_bf16.h>
#include <math.h>

typedef __attribute__((ext_vector_type(2))) float v2f;
typedef __attribute__((ext_vector_type(8))) float v8f;

#define T_DIM 4096
#define C_DIM 512
#define H_DIM 8
#define D_DIM 64
#define WIN   256

static __device__ __forceinline__ v8f wmma_f32(v2f a, v2f b, v8f c) {
  // D = A(16x4,f32) * B(4x16,f32) + C(16x16,f32), exact fp32 on the matrix pipe
  return __builtin_amdgcn_wmma_f32_16x16x4_f32(false, a, false, b, (short)0, c, false, false);
}

// D[M,N] = A[M,K] @ Bm[N,K]^T   (Linear layer without bias: x @ W^T)
// Each wave computes a 64x64 output block = 4x4 grid of 16x16 WMMA tiles.
// Per K-step of 4: 8 x b64 loads feed 16 WMMAs (4x register reuse vs 1 tile/wave).
__global__ __launch_bounds__(256) void wmma_gemm_nt(const float* __restrict__ A,
                                                    const float* __restrict__ Bm,
                                                    float* __restrict__ D,
                                                    int M, int N, int K) {
  const int lane  = threadIdx.x & 31;
  const int wave  = threadIdx.x >> 5;
  const int gwave = blockIdx.x * (blockDim.x >> 5) + wave;
  const int nBlocks = N >> 6;                 // 64-wide blocks
  const int mBlock = gwave / nBlocks;
  const int nBlock = gwave % nBlocks;
  if (mBlock * 64 >= M) return;               // uniform per wave -> EXEC all-ones for WMMA

  const int hl   = lane >> 4;                 // 16-lane half
  const int l16  = lane & 15;
  const int koff = hl * 2;                    // f32 A/B k sub-offset per ISA layout
  const int m0 = mBlock * 64, n0 = nBlock * 64;

  const float* arow[4];
  const float* brow[4];
  #pragma unroll
  for (int i = 0; i < 4; ++i)
    arow[i] = A + (size_t)(m0 + i * 16 + l16) * K + koff;   // A[m][kk+koff..+1]
  #pragma unroll
  for (int j = 0; j < 4; ++j)
    brow[j] = Bm + (size_t)(n0 + j * 16 + l16) * K + koff;  // B^T[kk+koff][n] = Bm[n][kk+koff]

  v8f acc[4][4];
  {
    v8f zero = {};
    #pragma unroll
    for (int i = 0; i < 4; ++i)
      #pragma unroll
      for (int j = 0; j < 4; ++j) acc[i][j] = zero;
  }

  #pragma unroll 2
  for (int kk = 0; kk < K; kk += 4) {
    v2f a[4], b[4];
    #pragma unroll
    for (int i = 0; i < 4; ++i) a[i] = *(const v2f*)(arow[i] + kk);
    #pragma unroll
    for (int j = 0; j < 4; ++j) b[j] = *(const v2f*)(brow[j] + kk);
    #pragma unroll
    for (int i = 0; i < 4; ++i)
      #pragma unroll
      for (int j = 0; j < 4; ++j)
        acc[i][j] = wmma_f32(a[i], b[j], acc[i][j]);
  }

  // C-layout store per tile: row = r + hl*8, col = l16
  #pragma unroll
  for (int i = 0; i < 4; ++i) {
    #pragma unroll
    for (int j = 0; j < 4; ++j) {
      float* dptr = D + (size_t)(m0 + i * 16 + hl * 8) * N + n0 + j * 16 + l16;
      #pragma unroll
      for (int r = 0; r < 8; ++r) dptr[(size_t)r * N] = acc[i][j][r];
    }
  }
}

// Sliding-window causal attention, flash-style online softmax.
// One wave per (batch, head, 16-query tile). qkv layout: [B, T, 3C], q|k|v contiguous.
// y layout: [B, T, C] with c = h*64 + d (matches reference transpose-back).
__global__ __launch_bounds__(256) void attn_sliding_window(const float* __restrict__ qkv,
                                                           float* __restrict__ y,
                                                           int Bsz) {
  __shared__ float ldsbuf[8 * 256];      // 1KB (16x16 f32) per wave for P transpose
  const int lane = threadIdx.x & 31;
  const int wave = threadIdx.x >> 5;
  const int gw   = blockIdx.x * (blockDim.x >> 5) + wave;
  const int qTiles = T_DIM / 16;
  const int qt = gw % qTiles;
  const int hb = gw / qTiles;
  const int h  = hb % H_DIM;
  const int b  = hb / H_DIM;
  if (b >= Bsz) return;

  const int hl   = lane >> 4;
  const int l16  = lane & 15;
  const int koff = hl * 2;
  const int q0   = qt * 16;
  const size_t row = 3 * C_DIM;

  // Preload Q tile (16 x 64) in WMMA A-layout: 16 v2f per lane
  const float* qbase = qkv + (size_t)(b * T_DIM + q0 + l16) * row + h * D_DIM + koff;
  v2f qa[16];
  #pragma unroll
  for (int kk = 0; kk < 16; ++kk) qa[kk] = *(const v2f*)(qbase + kk * 4);

  float mrow[8], lrow[8];
  v8f o[4];
  v8f zero = {};
  #pragma unroll
  for (int r = 0; r < 8; ++r) { mrow[r] = -__builtin_inff(); lrow[r] = 0.f; }
  #pragma unroll
  for (int nt = 0; nt < 4; ++nt) o[nt] = zero;

  float* pt = ldsbuf + wave * 256;
  const int tstart = (q0 >= WIN) ? 0 : ((WIN - q0) >> 4);  // skip fully-negative key tiles

  for (int t = tstart; t <= 16; ++t) {
    const int j0 = q0 - WIN + t * 16;

    // S = Q K^T : A = Q (16x64), B = K^T (64x16), 16 WMMA of K=4
    const float* kbase = qkv + (size_t)(b * T_DIM + j0 + l16) * row + C_DIM + h * D_DIM + koff;
    v8f s = {};
    #pragma unroll
    for (int kk = 0; kk < 16; ++kk) {
      v2f bv = *(const v2f*)(kbase + kk * 4);
      s = wmma_f32(qa[kk], bv, s);
    }

    // scale + sliding-window causal mask (only edge tiles actually mask anything)
    const int j = j0 + l16;
    float p[8];
    #pragma unroll
    for (int r = 0; r < 8; ++r) {
      const int i = q0 + r + hl * 8;
      const bool ok = (j <= i) && (i - j <= WIN);
      p[r] = ok ? s[r] * 0.125f : -__builtin_inff();   // 1/sqrt(64)
    }

    // online softmax: row stats via 16-lane xor reductions (wave32-safe)
    float corr[8];
    #pragma unroll
    for (int r = 0; r < 8; ++r) {
      float v = p[r];
      v = fmaxf(v, __shfl_xor(v, 1, 32));
      v = fmaxf(v, __shfl_xor(v, 2, 32));
      v = fmaxf(v, __shfl_xor(v, 4, 32));
      v = fmaxf(v, __shfl_xor(v, 8, 32));
      const float mnew = fmaxf(mrow[r], v);
      corr[r] = __expf(mrow[r] - mnew);
      mrow[r] = mnew;
      p[r] = __expf(p[r] - mnew);
      float rs = p[r];
      rs += __shfl_xor(rs, 1, 32);
      rs += __shfl_xor(rs, 2, 32);
      rs += __shfl_xor(rs, 4, 32);
      rs += __shfl_xor(rs, 8, 32);
      lrow[r] = lrow[r] * corr[r] + rs;
    }
    #pragma unroll
    for (int nt = 0; nt < 4; ++nt)
      #pragma unroll
      for (int r = 0; r < 8; ++r) o[nt][r] *= corr[r];

    // P: C-layout -> A-layout via wave-private LDS slice
    #pragma unroll
    for (int r = 0; r < 8; ++r) pt[(r + hl * 8) * 16 + l16] = p[r];
    asm volatile("s_wait_dscnt 0x0" ::: "memory");     // cross-lane DS store->load hazard
    v2f pa[4];
    #pragma unroll
    for (int kk = 0; kk < 4; ++kk)
      pa[kk] = *(const v2f*)(pt + l16 * 16 + kk * 4 + koff);

    // O += P V : A = P (16x16), B = V (16x64), 4 d-tiles x 4 K-steps
    const float* vbase = qkv + (size_t)(b * T_DIM + j0) * row + 2 * C_DIM + h * D_DIM;
    #pragma unroll
    for (int nt = 0; nt < 4; ++nt) {
      v8f acc = o[nt];
      #pragma unroll
      for (int kk = 0; kk < 4; ++kk) {
        const float* vp = vbase + (size_t)(kk * 4 + koff) * row + nt * 16 + l16;
        v2f bv = { vp[0], vp[row] };   // V[key][d], V[key+1][d]
        acc = wmma_f32(pa[kk], bv, acc);
      }
      o[nt] = acc;
    }
  }

  // normalize and store y[b, q0+m, h*64 + d]
  float* ybase = y + (size_t)(b * T_DIM + q0 + hl * 8) * C_DIM + h * D_DIM + l16;
  #pragma unroll
  for (int r = 0; r < 8; ++r) {
    const float inv = 1.0f / lrow[r];
    #pragma unroll
    for (int nt = 0; nt < 4; ++nt)
      ybase[(size_t)r * C_DIM + nt * 16] = o[nt][r] * inv;
  }
}

extern "C" void kernel_launch(void* const* d_in, const int* in_sizes, int n_in,
                              void* d_out, int out_size, void* d_ws, size_t ws_size,
                              hipStream_t stream) {
  const float* x      = (const float*)d_in[0];
  const float* w_attn = (const float*)d_in[1];
  const float* w_proj = (const float*)d_in[2];
  float* out = (float*)d_out;

  const int Bsz = in_sizes[0] / (T_DIM * C_DIM);
  const int M   = Bsz * T_DIM;

  float* qkv = (float*)d_ws;                       // [M, 3C]
  float* yy  = qkv + (size_t)M * 3 * C_DIM;        // [M, C]

  {  // 1) QKV projection: qkv = x @ w_attn^T   (M x 1536, K=512), 64x64 per wave
    const int waves = (M / 64) * ((3 * C_DIM) / 64);
    const int blocks = (waves + 7) / 8;
    wmma_gemm_nt<<<blocks, 256, 0, stream>>>(x, w_attn, qkv, M, 3 * C_DIM, C_DIM);
  }
  {  // 2) sliding-window attention
    const int waves = Bsz * H_DIM * (T_DIM / 16);
    const int blocks = (waves + 7) / 8;
    attn_sliding_window<<<blocks, 256, 0, stream>>>(qkv, yy, Bsz);
  }
  {  // 3) output projection: out = y @ w_proj^T  (M x 512, K=512), 64x64 per wave
    const int waves = (M / 64) * (C_DIM / 64);
    const int blocks = (waves + 7) / 8;
    wmma_gemm_nt<<<blocks, 256, 0, stream>>>(yy, w_proj, out, M, C_DIM, C_DIM);
  }
}